// AsyncTFCriterion_86698209837350
// MI455X (gfx1250) — compile-verified
//
#include <hip/hip_runtime.h>
#include <hip/hip_bf16.h>
#include <math.h>

// Problem constants (match reference setup_inputs()).
#define N_ 512
#define C_ 157
#define CP_ 160   // C padded to a multiple of 16 (and of the K-step 4)
#define M_ 20
#define SIGMA_ 300.0f
#define INV_DECAY (1.0f / 0.9f)
#define EPS_ 1e-7f

typedef __attribute__((ext_vector_type(2))) float v2f;
typedef __attribute__((ext_vector_type(8))) float v8f;

// ---------------------------------------------------------------------------
// Phase 1: per-(n) message gather. One block per n, 160 threads (c = channel).
// Each thread walks the M=20 bank entries keeping running decay weights for
// the past/future conditions (rank = # of prior valid entries -> running w).
// Also zero-initializes the loss accumulator (block 0, spare thread).
// ---------------------------------------------------------------------------
__global__ __launch_bounds__(160) void msg_kernel(
    const float* __restrict__ bank_values, const int* __restrict__ bank_times,
    const unsigned char* __restrict__ bank_mask, const int* __restrict__ ids,
    const int* __restrict__ times, float* __restrict__ msg,
    float* __restrict__ fmsg, float* __restrict__ loss_acc) {
  const int n = blockIdx.x;
  const int c = threadIdx.x;
  if (n == 0 && c == 159) *loss_acc = 0.0f;

  const int id = ids[n];
  const float t0 = (float)times[n];
  const float* vals = bank_values + (size_t)id * (M_ * C_);
  const int* tsp = bank_times + (size_t)id * M_;
  const unsigned char* mkp = bank_mask + (size_t)id * M_;

  const float inv2s2 = 1.0f / (2.0f * SIGMA_ * SIGMA_);
  float numP = 0.0f, denP = 0.0f, wP = 1.0f;
  float numF = 0.0f, denF = 0.0f, wF = 1.0f;
#pragma unroll
  for (int m = 0; m < M_; ++m) {
    const float ts = (float)tsp[m];
    const bool mk = mkp[m] != 0;
    const float d = ts - t0;
    const float kern = __expf(-d * d * inv2s2);
    const float v = (c < C_) ? vals[(size_t)m * C_ + c] : 0.0f;
    if (mk && ts < t0) { numP += wP * kern * v; denP += wP; wP *= INV_DECAY; }
    if (mk && ts > t0) { numF += wF * kern * v; denF += wF; wF *= INV_DECAY; }
  }
  if (c < C_) {
    msg[(size_t)n * C_ + c] = (denP > 0.0f) ? numP / fmaxf(denP, EPS_) : 0.0f;
    fmsg[(size_t)n * C_ + c] = (denF > 0.0f) ? numF / fmaxf(denF, EPS_) : 0.0f;
  }
}

// ---------------------------------------------------------------------------
// Phase 2: one block per n (10 waves, 320 threads). Wave w owns output strip
// [16w, 16w+16). Both bilinear terms run as (replicated row vector) x matrix
// on V_WMMA_F32_16X16X4_F32; all D rows are equal so lane nn's acc[0] is the
// strip value for column nn.
//   A layout (16x4 f32, 2 VGPRs): vgpr0 holds K = 2*half, vgpr1 K = 2*half+1
//   B layout (4x16 f32, 2 VGPRs): lane = N, same K-per-vgpr split as A.
// msg/fmsg are staged in LDS zero-padded to 160, so A-fragment loads are
// unconditional aligned ds_load_b64 and the K tail (157..159) contributes 0
// regardless of what the clamped B loads fetch. Control flow through the
// WMMA loop is uniform -> EXEC stays all-ones as the ISA requires.
// ---------------------------------------------------------------------------
__global__ __launch_bounds__(320) void qa_kernel(
    const float* __restrict__ a, const float* __restrict__ aa,
    const float* __restrict__ target, const float* __restrict__ msg,
    const float* __restrict__ fmsg, float* __restrict__ out,
    float* __restrict__ loss_acc) {
  __shared__ float s_msg[CP_];
  __shared__ float s_fmsg[CP_];
  __shared__ float s_sum;

  const int n = blockIdx.x;
  const int tid = threadIdx.x;
  if (tid == 0) s_sum = 0.0f;
  if (tid < CP_) {
    s_msg[tid] = (tid < C_) ? msg[(size_t)n * C_ + tid] : 0.0f;
  } else {
    const int u = tid - CP_;
    s_fmsg[u] = (u < C_) ? fmsg[(size_t)n * C_ + u] : 0.0f;
  }
  __syncthreads();

  const int strip = tid >> 5;  // wave id, 0..9
  const int lane = tid & 31;
  const int kh = lane >> 4;    // half-wave -> which K pair
  const int lp = lane & 15;    // N (column) index within strip

  const float* aaN = aa + (size_t)n * (C_ * C_);
  const int j0 = strip * 16;
  const int jc = min(j0 + lp, C_ - 1);  // clamped strip index (tail strip)
  const float* rowJ = aaN + (size_t)jc * C_;

  v8f acc1 = {0.f, 0.f, 0.f, 0.f, 0.f, 0.f, 0.f, 0.f};
  v8f acc2 = {0.f, 0.f, 0.f, 0.f, 0.f, 0.f, 0.f, 0.f};

#pragma unroll 4
  for (int k0 = 0; k0 < CP_; k0 += 4) {
    const int ka = k0 + 2 * kh;           // even -> 8B-aligned LDS b64 load
    const int kac = min(ka, C_ - 1);      // address-safe clamps; values for
    const int kbc = min(ka + 1, C_ - 1);  // k >= C_ are killed by A == 0

    const v2f A1 = *(const v2f*)&s_msg[ka];
    const v2f A2 = *(const v2f*)&s_fmsg[ka];

    // y1[j] += sum_k msg[k] * aa[k][j]   (coalesced row loads)
    v2f B1;
    B1.x = aaN[(size_t)kac * C_ + jc];
    B1.y = aaN[(size_t)kbc * C_ + jc];
    acc1 = __builtin_amdgcn_wmma_f32_16x16x4_f32(false, A1, false, B1,
                                                 (short)0, acc1, false, false);

    // y2[i] += sum_k fmsg[k] * aa[i][k]  (aa^T as the B operand)
    v2f B2;
    B2.x = rowJ[kac];
    B2.y = rowJ[kbc];
    acc2 = __builtin_amdgcn_wmma_f32_16x16x4_f32(false, A2, false, B2,
                                                 (short)0, acc2, false, false);
  }

  // All D rows identical -> lane lp's acc[0] is the strip value for index lp.
  const float y = acc1[0] + acc2[0];
  const int j = j0 + lane;  // lanes 16..31 duplicate lanes 0..15
  float local = 0.0f;
  if (lane < 16 && j < C_) {
    const float av = a[(size_t)n * C_ + j];
    const float q = 1.0f / (1.0f + __expf(-(av + y)));
    out[(size_t)n * C_ + j] = q;

    const float tv = target[(size_t)n * C_ + j];
    float p1 = fminf(fmaxf(q, EPS_), 1.0f - EPS_);
    local = tv * __logf(p1) + (1.0f - tv) * __logf(1.0f - p1);
    float p0 = 1.0f / (1.0f + __expf(-av));
    p0 = fminf(fmaxf(p0, EPS_), 1.0f - EPS_);
    local += tv * __logf(p0) + (1.0f - tv) * __logf(1.0f - p0);
  }
  // wave32 butterfly reduction; inactive lanes contribute 0
#pragma unroll
  for (int off = 16; off > 0; off >>= 1) {
    local += __shfl_xor(local, off, 32);
  }
  if (lane == 0) atomicAdd(&s_sum, local);
  __syncthreads();
  if (tid == 0) atomicAdd(loss_acc, s_sum);
}

// loss = (bce(qa) + bce(sigmoid(a))) / 3 = -(sum of both term-sums)/(N*C*3)
__global__ void finalize_kernel(const float* __restrict__ loss_acc,
                                float* __restrict__ out) {
  out[N_ * C_] = -loss_acc[0] / ((float)(N_ * C_) * 3.0f);
}

extern "C" void kernel_launch(void* const* d_in, const int* in_sizes, int n_in,
                              void* d_out, int out_size, void* d_ws,
                              size_t ws_size, hipStream_t stream) {
  (void)in_sizes; (void)n_in; (void)out_size; (void)ws_size;
  const float* a = (const float*)d_in[0];
  const float* aa = (const float*)d_in[1];
  const float* target = (const float*)d_in[2];
  const float* bank_values = (const float*)d_in[3];
  const int* bank_times = (const int*)d_in[4];
  const unsigned char* bank_mask = (const unsigned char*)d_in[5];
  const int* ids = (const int*)d_in[6];
  const int* times = (const int*)d_in[7];
  float* out = (float*)d_out;

  float* ws = (float*)d_ws;
  float* ws_loss = ws;                        // 1 float (padded to 64)
  float* ws_msg = ws + 64;                    // [512*157]
  float* ws_fmsg = ws_msg + (size_t)N_ * C_;  // [512*157]

  msg_kernel<<<N_, 160, 0, stream>>>(bank_values, bank_times, bank_mask, ids,
                                     times, ws_msg, ws_fmsg, ws_loss);
  qa_kernel<<<N_, 320, 0, stream>>>(a, aa, target, ws_msg, ws_fmsg, out,
                                    ws_loss);
  finalize_kernel<<<1, 1, 0, stream>>>(ws_loss, out);
}